// ContactLoss_89910845375012
// MI455X (gfx1250) — compile-verified
//
#include <hip/hip_runtime.h>
#include <math.h>

#define B_ 32
#define NH 778
#define NO 8192
#define NH_TILES 49            // ceil(778/16)
#define NHP (NH_TILES * 16)    // 784 padded rows
#define NO_TILES (NO / 16)     // 512
#define WAVES_PER_BLOCK 8
#define THRESH 0.025f          // CONTACT_THRESH == COLLISION_THRESH
#define BIGF 1e10f

typedef float v2f __attribute__((ext_vector_type(2)));
typedef float v8f __attribute__((ext_vector_type(8)));

// ---------------- workspace init (graph-capture safe, every launch) --------
__global__ void ws_init(unsigned* __restrict__ rowmin,
                        unsigned* __restrict__ colmin,
                        float* __restrict__ accum) {
  const int i = blockIdx.x * blockDim.x + threadIdx.x;
  const unsigned big = __float_as_uint(BIGF);
  if (i < B_ * NHP) rowmin[i] = big;
  if (i < B_ * NO)  colmin[i] = big;
  if (i < 8)        accum[i]  = 0.0f;
}

// ---------------- main WMMA pairwise-sqdist min kernel ---------------------
// D = A*B + C with A=(-2x,-2y,-2z,1) [16x4], B=(ox,oy,oz,|o|^2) [4x16],
// C=|h|^2 broadcast per row  ->  D[M,N] = squared distance.
// Software-pipelined: next tile's object vertices are loaded before the
// current tile's WMMA + min reduction, overlapping VMEM with matrix work.
__global__ void __launch_bounds__(256)
pairdist_min_wmma(const float* __restrict__ hand,
                  const float* __restrict__ obj,
                  const int* __restrict__ split,
                  unsigned* __restrict__ rowmin,
                  unsigned* __restrict__ colmin) {
  const int strip = blockIdx.x;              // b * NH_TILES + ht
  const int b    = strip / NH_TILES;
  const int ht   = strip - b * NH_TILES;
  const int wave = threadIdx.x >> 5;
  const int lane = threadIdx.x & 31;
  const int half = lane >> 4;                // 0: lanes 0-15, 1: lanes 16-31
  const int l16  = lane & 15;

  // ---- hand tile: build A operand and |h|^2 for the C broadcast ----
  const int hrow   = ht * 16 + l16;
  const int hclamp = hrow < NH ? hrow : NH - 1;
  const float* hp  = hand + ((size_t)b * NH + hclamp) * 3;
  float hx = hp[0], hy = hp[1], hz = hp[2];
  float hxx = (hrow < NH) ? (hx * hx + hy * hy + hz * hz) : BIGF;
  if (hrow >= NH) { hx = 0.0f; hy = 0.0f; hz = 0.0f; }

  v2f a;                                     // A 16x4 f32: K0,K1 | K2,K3 per half
  a.x = half ? (-2.0f * hz) : (-2.0f * hx);
  a.y = half ? 1.0f         : (-2.0f * hy);

  v8f c;                                     // C[r] = |h|^2 of row r + 8*half
#pragma unroll
  for (int r = 0; r < 8; ++r)
    c[r] = __shfl(hxx, r + (half << 3), 32);

  const int nvalid = split[b];

  float rmv[8];
#pragma unroll
  for (int r = 0; r < 8; ++r) rmv[r] = BIGF;

  const int tiles   = NO_TILES / WAVES_PER_BLOCK;   // 64 column tiles per wave
  const int colbase = wave * tiles * 16 + l16;
  const float* objb = obj + (size_t)b * NO * 3;

  // prologue: load tile 0
  float ox = objb[(size_t)colbase * 3 + 0];
  float oy = objb[(size_t)colbase * 3 + 1];
  float oz = objb[(size_t)colbase * 3 + 2];

  for (int t = 0; t < tiles; ++t) {
    const int col = colbase + t * 16;

    // stage current tile's data
    const float cx = ox, cy = oy, cz = oz;

    // prefetch next tile (clamped -> branch-free, in-bounds, EXEC stays full)
    const int nt   = (t + 1 < tiles) ? (t + 1) : t;
    const int ncol = colbase + nt * 16;
    ox = objb[(size_t)ncol * 3 + 0];
    oy = objb[(size_t)ncol * 3 + 1];
    oz = objb[(size_t)ncol * 3 + 2];

    float oyy = cx * cx + cy * cy + cz * cz;
    if (col >= nvalid) oyy = BIGF;           // padded object slots -> BIG dist

    v2f bm;                                  // B 4x16 f32: K0,K1 | K2,K3 per half
    bm.x = half ? cz  : cx;
    bm.y = half ? oyy : cy;

    v8f d = __builtin_amdgcn_wmma_f32_16x16x4_f32(
        false, a, false, bm, (short)0, c, false, false);

    float cm = BIGF;
#pragma unroll
    for (int r = 0; r < 8; ++r) {
      const float dv = d[r];
      rmv[r] = fminf(rmv[r], dv);            // running row-min across tiles
      cm     = fminf(cm, dv);                // column-min over 8 rows in half
    }
    cm = fminf(cm, __shfl_xor(cm, 16, 32));  // fold rows 0-7 with 8-15
    if (lane < 16)
      atomicMin(&colmin[(size_t)b * NO + col],
                __float_as_uint(fmaxf(cm, 0.0f)));
  }

  // ---- row-min reduction across the 16 lanes of each half ----
#pragma unroll
  for (int r = 0; r < 8; ++r) {
    float v = rmv[r];
    v = fminf(v, __shfl_xor(v, 1, 32));
    v = fminf(v, __shfl_xor(v, 2, 32));
    v = fminf(v, __shfl_xor(v, 4, 32));
    v = fminf(v, __shfl_xor(v, 8, 32));
    if (l16 == 0) {
      const int row = ht * 16 + r + (half << 3);
      if (row < NH)
        atomicMin(&rowmin[(size_t)b * NHP + row],
                  __float_as_uint(fmaxf(v, 0.0f)));
    }
  }
}

// ---------------- hand-side masked sums (missed + penetr) ------------------
__global__ void hand_accum(const unsigned* __restrict__ rowmin,
                           const int* __restrict__ ext_hand,
                           float* __restrict__ accum) {
  const int i = blockIdx.x * blockDim.x + threadIdx.x;
  float sm = 0.f, cmc = 0.f, sp = 0.f, cp = 0.f;
  if (i < B_ * NH) {
    const int b = i / NH;
    const int h = i - b * NH;
    const float dist = sqrtf(__uint_as_float(rowmin[(size_t)b * NHP + h]));
    const float val  = THRESH * tanhf(dist * (1.0f / THRESH));
    if (ext_hand[i] != 0) { sm = val; cmc = 1.f; }   // missed: exterior
    else                  { sp = val; cp  = 1.f; }   // penetr: ~exterior
  }
#pragma unroll
  for (int off = 16; off > 0; off >>= 1) {
    sm  += __shfl_down(sm,  off, 32);
    cmc += __shfl_down(cmc, off, 32);
    sp  += __shfl_down(sp,  off, 32);
    cp  += __shfl_down(cp,  off, 32);
  }
  if ((threadIdx.x & 31) == 0) {
    atomicAdd(&accum[0], sm);
    atomicAdd(&accum[1], cmc);
    atomicAdd(&accum[2], sp);
    atomicAdd(&accum[3], cp);
  }
}

// ---------------- object-side masked sum (penetr_o) ------------------------
__global__ void obj_accum(const unsigned* __restrict__ colmin,
                          const int* __restrict__ split,
                          const int* __restrict__ ext_obj,
                          float* __restrict__ accum) {
  const int i = blockIdx.x * blockDim.x + threadIdx.x;
  float s = 0.f, cnt = 0.f;
  if (i < B_ * NO) {
    const int b = i / NO;
    const int o = i - b * NO;
    const bool valid = o < split[b];
    if (valid && (ext_obj[i] == 0)) {
      const float dist = sqrtf(__uint_as_float(colmin[i]));
      s   = THRESH * tanhf(dist * (1.0f / THRESH));
      cnt = 1.f;
    }
  }
#pragma unroll
  for (int off = 16; off > 0; off >>= 1) {
    s   += __shfl_down(s,   off, 32);
    cnt += __shfl_down(cnt, off, 32);
  }
  if ((threadIdx.x & 31) == 0) {
    atomicAdd(&accum[4], s);
    atomicAdd(&accum[5], cnt);
  }
}

// ---------------- finalize: three masked means -----------------------------
__global__ void finalize(const float* __restrict__ accum, float* __restrict__ out) {
  const int k = threadIdx.x;
  if (blockIdx.x == 0 && k < 3) {
    const float num = accum[2 * k];
    const float den = accum[2 * k + 1];
    out[k] = (den > 0.0f) ? num / fmaxf(den, 1.0f) : 0.0f;
  }
}

extern "C" void kernel_launch(void* const* d_in, const int* in_sizes, int n_in,
                              void* d_out, int out_size, void* d_ws, size_t ws_size,
                              hipStream_t stream) {
  (void)in_sizes; (void)n_in; (void)out_size; (void)ws_size;

  const float* hand     = (const float*)d_in[0];   // [32, 778, 3] f32
  const float* obj      = (const float*)d_in[1];   // [32, 8192, 3] f32
  const int*   split    = (const int*)d_in[2];     // [32] i32
  const int*   ext_hand = (const int*)d_in[3];     // [32, 778]  bool->int
  const int*   ext_obj  = (const int*)d_in[4];     // [32, 8192] bool->int
  float* out = (float*)d_out;                      // [3] f32

  unsigned* rowmin = (unsigned*)d_ws;              // 32*784  u32
  unsigned* colmin = rowmin + (size_t)B_ * NHP;    // 32*8192 u32
  float*    accum  = (float*)(colmin + (size_t)B_ * NO);  // 8 f32

  // init mins to BIG, accumulators to 0
  {
    const int n = B_ * NO;                         // covers all three ranges
    ws_init<<<(n + 255) / 256, 256, 0, stream>>>(rowmin, colmin, accum);
  }
  // WMMA pairwise distance + row/col mins: one block per (batch, hand strip)
  pairdist_min_wmma<<<B_ * NH_TILES, 256, 0, stream>>>(hand, obj, split,
                                                       rowmin, colmin);
  // masked-mean accumulations
  hand_accum<<<(B_ * NH + 255) / 256, 256, 0, stream>>>(rowmin, ext_hand, accum);
  obj_accum<<<(B_ * NO + 255) / 256, 256, 0, stream>>>(colmin, split, ext_obj, accum);
  finalize<<<1, 32, 0, stream>>>(accum, out);
}